// MenuLoss_24172075942229
// MI455X (gfx1250) — compile-verified
//
#include <hip/hip_runtime.h>

#define N_COEFFS 447
#define N_IDS    223      // valid rounded ids: 0..222
#define INNER    7168     // 7*16*64
#define BATCH    512
#define TPB      256
#define EPT      28       // INNER / TPB

typedef float v2f __attribute__((ext_vector_type(2)));
typedef float v8f __attribute__((ext_vector_type(8)));

// ---------------------------------------------------------------------------
// 32-lane sum via the matrix pipe: D = A x Ones + C  (V_WMMA_F32_16X16X4_F32).
// Lane L supplies A[L%16][{0,1} or {2,3}] = (v, 0)  ->  row m of D holds
// v_m + v_{m+16} broadcast across N. Lanes<16 hold S_0..S_7 in the 8 D VGPRs,
// lanes>=16 hold S_8..S_15, so (intra-lane sum of 8) + shfl_xor(16) = total.
// Requires EXEC all-ones (guaranteed at call sites: no divergence).
// ---------------------------------------------------------------------------
__device__ __forceinline__ float wave_sum_wmma(float v) {
    v2f a; a[0] = v;    a[1] = 0.0f;
    v2f b; b[0] = 1.0f; b[1] = 1.0f;      // all-ones B: layout-independent
    v8f c = {0.0f, 0.0f, 0.0f, 0.0f, 0.0f, 0.0f, 0.0f, 0.0f};
    c = __builtin_amdgcn_wmma_f32_16x16x4_f32(false, a, false, b,
                                              (short)0, c, false, false);
    float t = c[0] + c[1] + c[2] + c[3] + c[4] + c[5] + c[6] + c[7];
    return t + __shfl_xor(t, 16, 32);
}

// ---------------------------------------------------------------------------
// Phase 1: pred-calorie lookup table. round(pred_id) is an integer k in
// [0,222]; mask(k) depends only on k, so cheb(mask(round(id))) has just 223
// distinct values. 223 threads each run the 447-term recurrence once.
// ---------------------------------------------------------------------------
__global__ void build_table_kernel(const float* __restrict__ coeffs,
                                   float* __restrict__ table) {
    __shared__ float sc[N_COEFFS];
    int tid = threadIdx.x;
    for (int i = tid; i < N_COEFFS; i += TPB) sc[i] = coeffs[i];
    __syncthreads();
    if (tid < N_IDS) {
        float k   = (float)tid;
        float z   = 50.0f * (222.5f - k);
        float sig = 1.0f / (1.0f + __expf(-z));   // sigmoid
        float m   = k * sig;                       // _mask(k)
        float xn  = m * (1.0f / 111.0f) - 1.0f;
        float x2  = 2.0f * xn;
        float Tm1 = 1.0f, Tn = xn;
        float acc = __builtin_fmaf(sc[1], xn, sc[0]);
        for (int i = 2; i < N_COEFFS; ++i) {
            float Tnew = __builtin_fmaf(x2, Tn, -Tm1);
            acc = __builtin_fmaf(sc[i], Tnew, acc);
            Tm1 = Tn; Tn = Tnew;
        }
        table[tid] = acc;
    }
}

// ---------------------------------------------------------------------------
// Phase 2: one block per batch row. Each thread owns 28 elements, processed
// in groups of 4 so the Chebyshev recurrence runs 4 independent chains
// (8 independent FMAs per coefficient step) to cover FMA latency.
// Row partials reduced per-wave on the matrix pipe, then across 8 waves.
// ---------------------------------------------------------------------------
__global__ __launch_bounds__(TPB)
void menu_row_kernel(const float* __restrict__ y_pred,
                     const float* __restrict__ y,
                     const float* __restrict__ coeffs,
                     const float* __restrict__ table,
                     float* __restrict__ row_true,
                     float* __restrict__ row_pred,
                     float* __restrict__ row_pen) {
    __shared__ float sc[N_COEFFS];
    __shared__ float st[N_IDS];
    __shared__ float sred[24];

    const int b   = blockIdx.x;
    const int tid = threadIdx.x;

    for (int i = tid; i < N_COEFFS; i += TPB) sc[i] = coeffs[i];
    for (int i = tid; i < N_IDS;    i += TPB) st[i] = table[i];
    __syncthreads();

    const float2* yp2 = (const float2*)y_pred + (size_t)b * INNER;
    const float2* yt2 = (const float2*)y      + (size_t)b * INNER;

    float tp = 0.0f;   // sum(true_cal * true_amt)
    float pp = 0.0f;   // sum(pred_cal * pred_amt)
    float pen = 0.0f;  // zeros/nonzeros penalty + id range penalty

    #pragma unroll 1
    for (int g = 0; g < EPT / 4; ++g) {
        float x2v[4], Tm1[4], Tn[4], acc[4], tamt[4];
        #pragma unroll
        for (int e = 0; e < 4; ++e) {
            const int j = (g * 4 + e) * TPB + tid;   // coalesced
            float2 vp = yp2[j];
            float2 vt = yt2[j];
            const float pid  = vp.x, pamt = vp.y;
            const float tidv = vt.x;
            tamt[e] = vt.y;

            // ---- pred side: penalties + table lookup ----
            float th_id = tanhf(4.0f * pid);
            float th_am = tanhf(4.0f * pamt);
            float izm   = 1.0f - th_id;              // id_zero_mask
            pen += izm * th_am + th_id * (1.0f - th_am);   // case1 + case2
            pen += fmaxf(pid - 222.0f, 0.0f);              // relu range
            int kid = (int)rintf(pid);               // round-half-even
            kid = kid < 0 ? 0 : (kid > (N_IDS - 1) ? (N_IDS - 1) : kid);
            pp = __builtin_fmaf(st[kid], pamt, pp);

            // ---- true side: init recurrence ----
            float xn = tidv * (1.0f / 111.0f) - 1.0f;
            x2v[e] = 2.0f * xn;
            Tm1[e] = 1.0f; Tn[e] = xn;
            acc[e] = __builtin_fmaf(sc[1], xn, sc[0]);
        }

        // 445 steps, 4 independent chains interleaved (8 indep FMAs / step)
        #pragma unroll 5
        for (int i = 2; i < N_COEFFS; ++i) {
            const float c = sc[i];
            #pragma unroll
            for (int e = 0; e < 4; ++e) {
                float Tnew = __builtin_fmaf(x2v[e], Tn[e], -Tm1[e]);
                acc[e] = __builtin_fmaf(c, Tnew, acc[e]);
                Tm1[e] = Tn[e]; Tn[e] = Tnew;
            }
        }
        #pragma unroll
        for (int e = 0; e < 4; ++e)
            tp = __builtin_fmaf(acc[e], tamt[e], tp);
    }

    // per-wave reductions on the matrix pipe (EXEC all-ones here)
    float wtp  = wave_sum_wmma(tp);
    float wpp  = wave_sum_wmma(pp);
    float wpen = wave_sum_wmma(pen);

    const int lane = tid & 31, wave = tid >> 5;
    if (lane == 0) { sred[wave] = wtp; sred[8 + wave] = wpp; sred[16 + wave] = wpen; }
    __syncthreads();
    if (tid == 0) {
        float a = 0.0f, p = 0.0f, q = 0.0f;
        #pragma unroll
        for (int w = 0; w < 8; ++w) { a += sred[w]; p += sred[8 + w]; q += sred[16 + w]; }
        row_true[b] = a; row_pred[b] = p; row_pen[b] = q;
    }
}

// ---------------------------------------------------------------------------
// Phase 3: single wave. val_b = ((rt_b - rp_b)/700)^2 + pen_b, summed over
// 512 rows via 8 accumulating WMMAs; result = sum / 512 (batch mean).
// ---------------------------------------------------------------------------
__global__ void final_kernel(const float* __restrict__ row_true,
                             const float* __restrict__ row_pred,
                             const float* __restrict__ row_pen,
                             float* __restrict__ out) {
    const int L = threadIdx.x;                 // 32 threads = 1 wave
    v2f bones; bones[0] = 1.0f; bones[1] = 1.0f;
    v8f accD = {0.0f, 0.0f, 0.0f, 0.0f, 0.0f, 0.0f, 0.0f, 0.0f};
    const float s = 1.0f / 700.0f;             // (/100)/7 menu scaling

    #pragma unroll
    for (int it = 0; it < BATCH / 64; ++it) {
        const int i0 = it * 64 + L;
        const int i1 = i0 + 32;
        float d0 = (row_true[i0] - row_pred[i0]) * s;
        float d1 = (row_true[i1] - row_pred[i1]) * s;
        float v0 = __builtin_fmaf(d0, d0, row_pen[i0]);
        float v1 = __builtin_fmaf(d1, d1, row_pen[i1]);
        v2f a; a[0] = v0; a[1] = v1;
        accD = __builtin_amdgcn_wmma_f32_16x16x4_f32(false, a, false, bones,
                                                     (short)0, accD, false, false);
    }
    float t = accD[0] + accD[1] + accD[2] + accD[3] +
              accD[4] + accD[5] + accD[6] + accD[7];
    float total = t + __shfl_xor(t, 16, 32);
    if (L == 0) out[0] = total * (1.0f / (float)BATCH);
}

// ---------------------------------------------------------------------------
extern "C" void kernel_launch(void* const* d_in, const int* in_sizes, int n_in,
                              void* d_out, int out_size, void* d_ws, size_t ws_size,
                              hipStream_t stream) {
    (void)in_sizes; (void)n_in; (void)out_size; (void)ws_size;
    const float* y_pred = (const float*)d_in[0];
    const float* y      = (const float*)d_in[1];
    const float* coeffs = (const float*)d_in[2];

    float* ws       = (float*)d_ws;
    float* table    = ws;                    // 223 floats (pad to 256)
    float* row_true = ws + 256;              // 512
    float* row_pred = ws + 256 + 512;        // 512
    float* row_pen  = ws + 256 + 1024;       // 512   (total 7 KB)

    build_table_kernel<<<1, TPB, 0, stream>>>(coeffs, table);
    menu_row_kernel<<<BATCH, TPB, 0, stream>>>(y_pred, y, coeffs, table,
                                               row_true, row_pred, row_pen);
    final_kernel<<<1, 32, 0, stream>>>(row_true, row_pred, row_pen, (float*)d_out);
}